// SpectralConv2d_81716047773933
// MI455X (gfx1250) — compile-verified
//
#include <hip/hip_runtime.h>

typedef __attribute__((ext_vector_type(2))) float v2f;
typedef __attribute__((ext_vector_type(8))) float v8f;

// Problem constants (match reference)
#define IN_CH  32
#define OUT_CH 32
#define M1     16
#define M2     16
#define B_     8
#define L_     16
#define H_     64
#define W_     33

// Strides in floats
#define XW_C   (H_ * W_)          // 2112   x: c stride
#define XW_BL  (IN_CH * XW_C)     // 67584  x: flattened (b,l) stride
#define WS_C   (OUT_CH * M1 * M2) // 8192   w: c_in stride
#define WS_O   (M1 * M2)          // 256    w: c_out stride
#define OUT_O  (H_ * W_ * 2)      // 4224   out: o stride (trailing re/im dim)
#define OUT_BL (OUT_CH * OUT_O)   // 135168 out: (b,l) stride

// ---------------------------------------------------------------------------
// Kernel 1: zero-fill the whole output with wide b128 stores.
// This is the bandwidth-dominant part (~69 MB, mostly zeros by construction).
// ---------------------------------------------------------------------------
__global__ __launch_bounds__(256)
void spectral_zero_kernel(float* __restrict__ out, int n) {
    const int tid    = blockIdx.x * blockDim.x + threadIdx.x;
    const int stride = gridDim.x * blockDim.x;
    const int n4     = n >> 2;
    float4* __restrict__ out4 = (float4*)out;
    const float4 z = make_float4(0.f, 0.f, 0.f, 0.f);
    for (int idx = tid; idx < n4; idx += stride) out4[idx] = z;
    // tail (n is divisible by 4 here, but stay safe)
    for (int idx = (n4 << 2) + tid; idx < n; idx += stride) out[idx] = 0.f;
}

// ---------------------------------------------------------------------------
// Kernel 2: complex channel-mix GEMM per mode position via f32 WMMA.
//
// Wave work unit: (region, i, j, mchunk, nsplit) -> 2*16*16*2*4 = 4096 waves.
//   A = W[region][:, obase:obase+16, i, j]   (M=16 outch x K=32 inch, complex)
//       loaded once per wave, reused across both bl-chunks.
//   B = X[blbase:blbase+16, :, isrc, j]      (K=32 inch x N=16 bl, complex)
//   D = A x B accumulated over K in 8 steps of V_WMMA_F32_16X16X4_F32.
// Complex product via 4 accumulators: re = RR - II, im = RI + IR.
//
// WMMA f32 16x16x4 VGPR layouts (ISA 7.12.2):
//   A: lane m = lane&15 holds row M=m; VGPR0/1 = K = kk, kk+1 (kk = 2*(lane>=16))
//   B: lane n = lane&15 holds col N=n; VGPR0/1 = K = kk, kk+1
//   C/D: lane n = lane&15 is column; VGPR v holds row M = v + 8*(lane>=16)
// ---------------------------------------------------------------------------
__global__ __launch_bounds__(256)
void spectral_wmma_kernel(const float* __restrict__ xre, const float* __restrict__ xim,
                          const float* __restrict__ w1re, const float* __restrict__ w1im,
                          const float* __restrict__ w2re, const float* __restrict__ w2im,
                          float* __restrict__ out) {
    const int lane = threadIdx.x & 31;
    const int wid  = blockIdx.x * 8 + (threadIdx.x >> 5);  // 0..4095, uniform per wave

    const int ns     = wid & 3;         // bl-split: this wave covers 2 of 8 bl-chunks
    const int mchunk = (wid >> 2) & 1;  // out-channel half
    const int j      = (wid >> 3) & 15;
    const int i      = (wid >> 7) & 15;
    const int region = (wid >> 11) & 1;

    const float* __restrict__ wre = region ? w2re : w1re;
    const float* __restrict__ wim = region ? w2im : w1im;
    const int isrc = region ? (H_ - M1 + i) : i;  // bottom block reads/writes rows 48..63

    const int mn = lane & 15;            // M (for A/D rows...) / N column index per layout
    const int kk = (lane >> 4) << 1;     // 0 or 2: K sub-index held by this half-wave

    const int obase = mchunk * 16;
    const int o     = obase + mn;        // this lane's A row (out channel)
    const int ij_w  = i * M2 + j;
    const int ij_x  = isrc * W_ + j;

    // ---- Load A (16 x 32 complex weights) into registers, reused across bl-chunks.
    v2f a_re[8], a_im[8];
    const size_t wbase = (size_t)o * WS_O + ij_w;
#pragma unroll
    for (int s = 0; s < 8; ++s) {
        const int c0 = s * 4 + kk;
        a_re[s].x = wre[(size_t)c0 * WS_C + wbase];
        a_re[s].y = wre[(size_t)(c0 + 1) * WS_C + wbase];
        a_im[s].x = wim[(size_t)c0 * WS_C + wbase];
        a_im[s].y = wim[(size_t)(c0 + 1) * WS_C + wbase];
    }

#pragma unroll
    for (int t = 0; t < 2; ++t) {
        const int nbase = (ns * 2 + t) * 16;
        const int bl    = nbase + mn;                       // this lane's B/D column
        const size_t xbase = (size_t)bl * XW_BL + ij_x;

        // prefetch the far end of this chunk's x stream (gfx1250 global_prefetch_b8)
        __builtin_prefetch(&xre[xbase + (size_t)16 * XW_C], 0, 3);

        v8f rr = {}, ii = {}, ri = {}, ir = {};
#pragma unroll
        for (int s = 0; s < 8; ++s) {
            const int c0 = s * 4 + kk;
            v2f b_re, b_im;
            b_re.x = xre[xbase + (size_t)c0 * XW_C];
            b_re.y = xre[xbase + (size_t)(c0 + 1) * XW_C];
            b_im.x = xim[xbase + (size_t)c0 * XW_C];
            b_im.y = xim[xbase + (size_t)(c0 + 1) * XW_C];
            rr = __builtin_amdgcn_wmma_f32_16x16x4_f32(false, a_re[s], false, b_re,
                                                       (short)0, rr, false, false);
            ii = __builtin_amdgcn_wmma_f32_16x16x4_f32(false, a_im[s], false, b_im,
                                                       (short)0, ii, false, false);
            ri = __builtin_amdgcn_wmma_f32_16x16x4_f32(false, a_re[s], false, b_im,
                                                       (short)0, ri, false, false);
            ir = __builtin_amdgcn_wmma_f32_16x16x4_f32(false, a_im[s], false, b_re,
                                                       (short)0, ir, false, false);
        }
        const v8f ore = rr - ii;
        const v8f oim = ri + ir;

        // ---- Store: lane owns column bl; VGPR v is row (out channel) v + 8*hi.
        const int hi = lane >> 4;
        const size_t obrow = (size_t)bl * OUT_BL + (size_t)isrc * (W_ * 2) + (size_t)j * 2;
#pragma unroll
        for (int v = 0; v < 8; ++v) {
            const int m = v + 8 * hi;
            float2 val;
            val.x = ore[v];
            val.y = oim[v];
            *(float2*)(out + obrow + (size_t)(obase + m) * OUT_O) = val;  // 8B (re,im) pair
        }
    }
}

// ---------------------------------------------------------------------------
extern "C" void kernel_launch(void* const* d_in, const int* in_sizes, int n_in,
                              void* d_out, int out_size, void* d_ws, size_t ws_size,
                              hipStream_t stream) {
    const float* xre  = (const float*)d_in[0];
    const float* xim  = (const float*)d_in[1];
    const float* w1re = (const float*)d_in[2];
    const float* w1im = (const float*)d_in[3];
    const float* w2re = (const float*)d_in[4];
    const float* w2im = (const float*)d_in[5];
    float* out = (float*)d_out;

    // 1) zero-fill output (bandwidth-dominant)
    int blocks = (out_size / 4 + 255) / 256;
    if (blocks > 4096) blocks = 4096;
    if (blocks < 1) blocks = 1;
    spectral_zero_kernel<<<blocks, 256, 0, stream>>>(out, out_size);

    // 2) mode-region complex GEMMs via WMMA: 4096 waves = 512 blocks x 8 waves
    spectral_wmma_kernel<<<512, 256, 0, stream>>>(xre, xim, w1re, w1im,
                                                  w2re, w2im, out);
}